// LocalResponseNorm_17420387352942
// MI455X (gfx1250) — compile-verified
//
#include <hip/hip_runtime.h>

// ---------------------------------------------------------------------------
// Spatial LRN 5x5 for x:(16,96,224,224) f32.
//   out = x / (2 + 1e-4 * boxsum5x5(x^2))^0.75
// Strategy: per-block 56-row strip of one (n,c) plane. TDM (tensor_load_to_lds)
// DMAs a 60x224 halo tile into LDS; compute is a separable box filter:
// horizontal 5-sums from LDS (ds_load_b128), vertical 5-sum via register ring
// sliding down rows; pow via v_log_f32/v_exp_f32.
// ---------------------------------------------------------------------------

#define W         224
#define H         224
#define NPLANES   (16 * 96)
#define TILE_OUT  56                 // output rows per block
#define TILE_LDS  (TILE_OUT + 4)     // 60 rows incl. halo
#define NSUB      4                  // row sub-strips per block
#define SUB_ROWS  (TILE_OUT / NSUB)  // 14
#define NCG       (W / 4)            // 56 column groups of 4

typedef unsigned int u32x4 __attribute__((ext_vector_type(4)));
typedef int          i32x4 __attribute__((ext_vector_type(4)));
typedef int          i32x8 __attribute__((ext_vector_type(8)));

// Fast hw transcendentals: v_log_f32 is base-2 log, v_exp_f32 is exp2.
__device__ __forceinline__ float fast_log2(float v) {
  return __builtin_amdgcn_logf(v);
}
__device__ __forceinline__ float fast_exp2(float v) {
  return __builtin_amdgcn_exp2f(v);
}

// Horizontal sliding 5-window sum of squares for 4 consecutive columns.
// Loads 12 floats (cols c0-4 .. c0+7) as three b128 LDS loads with edge guards.
__device__ __forceinline__ void row_hsum(const float* __restrict__ tile, int lr,
                                         int cg, float hq[4], float4& mid) {
  const float4* rp = (const float4*)(tile + lr * W);
  float4 zz; zz.x = zz.y = zz.z = zz.w = 0.0f;
  float4 a = (cg > 0)       ? rp[cg - 1] : zz;
  float4 b = rp[cg];
  float4 c = (cg < NCG - 1) ? rp[cg + 1] : zz;
  float s2 = a.z * a.z, s3 = a.w * a.w;
  float s4 = b.x * b.x, s5 = b.y * b.y, s6 = b.z * b.z, s7 = b.w * b.w;
  float s8 = c.x * c.x, s9 = c.y * c.y;
  float m = s4 + s5 + s6;
  hq[0] = s2 + s3 + m;
  hq[1] = s3 + m + s7;
  hq[2] = m + s7 + s8;
  hq[3] = s5 + s6 + s7 + s8 + s9;
  mid = b;  // raw x at cols c0..c0+3 of this row
}

__global__ __launch_bounds__(256) void lrn5x5_kernel(const float* __restrict__ x,
                                                     float* __restrict__ out) {
  __shared__ __align__(16) float tile[TILE_LDS * W];  // 53,760 B

  const int plane = blockIdx.x >> 2;
  const int strip = blockIdx.x & 3;
  const int r0    = strip * TILE_OUT;

  const float* src = x   + (long long)plane * (H * W);
  float*       dst = out + (long long)plane * (H * W);

  const int row_lo = r0 - 2;
  const int start  = (row_lo < 0) ? 0 : row_lo;       // first global row DMA'd
  const int skip   = start - row_lo;                  // 0, or 2 at top edge
  const int nrows  = TILE_LDS - skip;                 // tile rows to DMA
  const int valid  = H - start;                       // tensor_dim1 (OOB rows -> 0)

  const int t = threadIdx.x;

  // Zero the 2 top-halo rows (only at the top edge of the plane).
  if (strip == 0) {
    if (t < W) {
      tile[t]     = 0.0f;
      tile[W + t] = 0.0f;
    }
  }

  const int wid = __builtin_amdgcn_readfirstlane((int)(threadIdx.x >> 5));
  if (wid == 0) {
    // ---- Tensor DMA descriptor (CDNA5 ISA ch.8) ----
    unsigned long long gaddr =
        (unsigned long long)(const void*)(src + (long long)start * W);
    // Low 32 bits of a flat LDS pointer are the LDS byte offset.
    unsigned lds_addr =
        (unsigned)(unsigned long long)(const void*)&tile[skip * W];

    u32x4 g0;
    g0[0] = 1u;                                           // count=1, is_restore=0
    g0[1] = lds_addr;                                     // lds_addr [63:32]
    g0[2] = (unsigned)(gaddr & 0xFFFFFFFFull);            // global_addr lo
    g0[3] = (unsigned)((gaddr >> 32) & 0x01FFFFFFull)     // global_addr hi
          | 0x80000000u;                                  // type=2 ("image")

    i32x8 g1;
    g1[0] = (2 << 16);                   // data_size=4B; wg_mask=0; no pad/iter
    g1[1] = (int)((unsigned)W << 16);    // tensor_dim0 lo16 @ [63:48]
    g1[2] = (int)((unsigned)valid << 16);// tensor_dim0 hi16=0; tensor_dim1 lo16
    g1[3] = (int)((unsigned)W << 16);    // tensor_dim1 hi16=0; tile_dim0=224
    g1[4] = nrows;                       // tile_dim1; tile_dim2=0
    g1[5] = W;                           // tensor_dim0_stride lo32
    g1[6] = 0;                           // stride0 hi16=0; stride1 lo16=0
    g1[7] = 0;                           // stride1 hi32=0
    i32x4 g2 = {0, 0, 0, 0};             // 2D tensor: groups 2/3 unused
    i32x4 g3 = {0, 0, 0, 0};

#if defined(__clang_major__) && (__clang_major__ >= 23)
    i32x8 g4 = {0, 0, 0, 0, 0, 0, 0, 0};
    __builtin_amdgcn_tensor_load_to_lds(g0, g1, g2, g3, g4, 0);
#else
    __builtin_amdgcn_tensor_load_to_lds(g0, g1, g2, g3, 0);
#endif
    __builtin_amdgcn_s_wait_tensorcnt(0);
  }
  __syncthreads();

  if (t < NSUB * NCG) {
    const int cg    = t % NCG;          // column group -> cols 4*cg..4*cg+3
    const int sub   = t / NCG;          // row sub-strip
    const int c0    = cg * 4;
    const int lbase = sub * SUB_ROWS;   // first LDS-local output row
    const int gr0   = r0 + lbase;       // first global output row

    float  h[5][4];                     // ring of horizontal sums
    float4 xm[2];                       // delayed center-row x values

    // Warm-up: rows lbase .. lbase+3.
#pragma unroll
    for (int k = 0; k < 4; ++k) {
      float4 mid;
      row_hsum(tile, lbase + k, cg, h[k], mid);
      if (k >= 2) xm[k & 1] = mid;      // rows lbase+2, lbase+3
    }

#pragma unroll
    for (int i = 0; i < SUB_ROWS; ++i) {
      float hn[4];
      float4 mid;
      row_hsum(tile, lbase + i + 4, cg, hn, mid);
#pragma unroll
      for (int k = 0; k < 4; ++k) h[(i + 4) % 5][k] = hn[k];

      float4 xa = xm[i & 1];            // x at output row (loaded 2 iters ago)
      const float xv[4] = {xa.x, xa.y, xa.z, xa.w};
      float4 ov;
#pragma unroll
      for (int k = 0; k < 4; ++k) {
        float s  = h[0][k] + h[1][k] + h[2][k] + h[3][k] + h[4][k];
        float dn = 2.0f + 1.0e-4f * s;                 // K + alpha * ssq
        float r  = fast_exp2(-0.75f * fast_log2(dn));  // dn^-beta
        ((float*)&ov)[k] = xv[k] * r;
      }
      xm[i & 1] = mid;                  // row lbase+i+4, consumed at iter i+2

      *(float4*)(dst + (long long)(gr0 + i) * W + c0) = ov;
    }
  }
}

extern "C" void kernel_launch(void* const* d_in, const int* in_sizes, int n_in,
                              void* d_out, int out_size, void* d_ws, size_t ws_size,
                              hipStream_t stream) {
  (void)in_sizes; (void)n_in; (void)d_ws; (void)ws_size; (void)out_size;
  const float* x = (const float*)d_in[0];
  float* out     = (float*)d_out;
  dim3 grid(NPLANES * 4);   // 6144 blocks: (plane << 2) | strip
  dim3 block(256);          // 8 wave32 waves; 224 compute lanes
  lrn5x5_kernel<<<grid, block, 0, stream>>>(x, out);
}